// PointNetFeaturePropagation_32512902431505
// MI455X (gfx1250) — compile-verified
//
#include <hip/hip_runtime.h>
#include <math.h>

// Problem constants (match reference)
#define B_    8
#define N_    16384
#define S_    2048
#define C1_   128
#define C2_   256
#define CIN1  (C1_ + C2_)   // 384
#define COUT  256
#define M_    (B_ * N_)     // 131072 rows
#define NBLK  (M_ / 16)     // 8192 GEMM row-tile blocks
#define BN_EPS 1e-5f
#define IDW_EPS 1e-8f

typedef __attribute__((ext_vector_type(2))) float v2f;
typedef __attribute__((ext_vector_type(8))) float v8f;
typedef __attribute__((ext_vector_type(4))) int   v4i;

// ---------------------------------------------------------------------------
// CDNA5 async global->LDS DMA (ASYNCcnt-tracked), with portable fallback.
// Probe-revealed builtin signature:
//   void __builtin_amdgcn_global_load_async_to_lds_b128(
//       v4i addrspace(1)* gsrc, v4i addrspace(3)* lds_dst, int offset, int cpol)
// ---------------------------------------------------------------------------
#if defined(__gfx1250__) && __has_builtin(__builtin_amdgcn_global_load_async_to_lds_b128)
#define HAVE_ASYNC_LDS 1
#else
#define HAVE_ASYNC_LDS 0
#endif

__device__ __forceinline__ void async_copy_b128(void* lds_dst, const void* gsrc)
{
#if HAVE_ASYNC_LDS
    __builtin_amdgcn_global_load_async_to_lds_b128(
        (__attribute__((address_space(1))) v4i*)(void*)gsrc,
        (__attribute__((address_space(3))) v4i*)lds_dst,
        /*offset=*/0, /*cpol=*/0);
#else
    *(float4*)lds_dst = *(const float4*)gsrc;
#endif
}

__device__ __forceinline__ void wait_async0()
{
#if HAVE_ASYNC_LDS
#if __has_builtin(__builtin_amdgcn_s_wait_asynccnt)
    __builtin_amdgcn_s_wait_asynccnt(0);
#else
    asm volatile("s_wait_asynccnt 0x0" ::: "memory");
#endif
#endif
}

// ---------------------------------------------------------------------------
// Kernel 1: 3-NN search.  One thread per query point; xyz2 slab of this batch
// DMA'd into LDS (2048*3*4 = 24KB, 1536 b128 transfers).  Top-3 by squared
// distance (same ordering as euclidean); IDW weights normalized at the end.
// ---------------------------------------------------------------------------
__global__ void knn_kernel(const float* __restrict__ xyz1,
                           const float* __restrict__ xyz2,
                           int* __restrict__ knn_idx,
                           float* __restrict__ knn_w)
{
    __shared__ float s_xyz[S_ * 3];
    const int blocksPerBatch = N_ / 256;              // 64
    const int b  = blockIdx.x / blocksPerBatch;
    const int n0 = (blockIdx.x % blocksPerBatch) * 256;
    const float* x2 = xyz2 + (size_t)b * S_ * 3;
    for (int i = threadIdx.x; i < (S_ * 3) / 4; i += 256)
        async_copy_b128(&s_xyz[i * 4], x2 + (size_t)i * 4);

    // overlap: fetch this thread's query point while the DMA runs
    const size_t m = (size_t)b * N_ + (n0 + threadIdx.x);
    const float qx = xyz1[m * 3 + 0];
    const float qy = xyz1[m * 3 + 1];
    const float qz = xyz1[m * 3 + 2];

    wait_async0();
    __syncthreads();

    float d0 = 3.4e38f, d1 = 3.4e38f, d2v = 3.4e38f;
    int   i0 = 0, i1 = 0, i2 = 0;
    for (int s = 0; s < S_; ++s) {
        float dx = qx - s_xyz[s * 3 + 0];
        float dy = qy - s_xyz[s * 3 + 1];
        float dz = qz - s_xyz[s * 3 + 2];
        float d  = dx * dx + dy * dy + dz * dz;
        if (d < d0)       { d2v = d1; i2 = i1; d1 = d0; i1 = i0; d0 = d; i0 = s; }
        else if (d < d1)  { d2v = d1; i2 = i1; d1 = d;  i1 = s; }
        else if (d < d2v) { d2v = d;  i2 = s; }
    }
    float w0 = 1.f / (sqrtf(d0)  + IDW_EPS);
    float w1 = 1.f / (sqrtf(d1)  + IDW_EPS);
    float w2 = 1.f / (sqrtf(d2v) + IDW_EPS);
    float inv = 1.f / (w0 + w1 + w2);
    knn_idx[m * 3 + 0] = i0; knn_idx[m * 3 + 1] = i1; knn_idx[m * 3 + 2] = i2;
    knn_w[m * 3 + 0] = w0 * inv;
    knn_w[m * 3 + 1] = w1 * inv;
    knn_w[m * 3 + 2] = w2 * inv;
}

// ---------------------------------------------------------------------------
// Shared fp32 WMMA GEMM core (ISA 7.12.2 layouts for 16x16x4 f32):
//   A 16x4 : VGPR0 = K=kb, VGPR1 = K=kb+1, kb = 2*(lane>=16), M = lane&15
//   B 4x16 : same K split, N = lane&15
//   C/D    : VGPR r -> row 8*(lane>=16)+r, col = lane&15
// 512 threads = 16 wave32s per block; block owns one 16-row M tile (A in LDS,
// row stride padded to K+4 floats -> conflict-free ds_load_2addr_b64);
// wave w owns output columns [16w,16w+16).  EXEC all-ones around every WMMA.
// Epilogue writes pre-BN output and deterministic per-block channel partials.
// ---------------------------------------------------------------------------
template <int K>
__device__ __forceinline__ void gemm_core_and_epilogue(
    const float* __restrict__ sA,        // LDS tile, row stride K+4
    const float* __restrict__ W,         // COUT x K row-major (L2-resident)
    const float* __restrict__ bias,
    float* __restrict__ Hout,            // M x COUT, pre-BN
    float* __restrict__ partial,         // [2*NBLK][COUT]: sums then sumsqs
    size_t mBase)
{
    constexpr int LDA = K + 4;
    const int wave = threadIdx.x >> 5;
    const int lane = threadIdx.x & 31;
    const int half = lane >> 4;
    const int l16  = lane & 15;
    const int o    = wave * 16 + l16;
    const int kb   = half * 2;

    v8f acc = {};
    const float* wrow = W + (size_t)o * K;
    const float* arow = sA + (size_t)l16 * LDA + kb;
    for (int k0 = 0; k0 < K; k0 += 4) {
        v2f a, bm;
        a.x  = arow[k0];
        a.y  = arow[k0 + 1];
        bm.x = wrow[k0 + kb];
        bm.y = wrow[k0 + kb + 1];
        acc = __builtin_amdgcn_wmma_f32_16x16x4_f32(
            false, a, false, bm, (short)0, acc, false, false);
    }

    const float bv = bias[o];
    float* out = Hout + (mBase + (size_t)half * 8) * COUT + o;
    float s = 0.f, s2 = 0.f;
#pragma unroll
    for (int r = 0; r < 8; ++r) {
        float v = acc[r] + bv;
        out[(size_t)r * COUT] = v;
        s  += v;
        s2 += v * v;
    }
    // combine the two half-waves holding the same channel (rows 0-7 / 8-15)
    s  += __shfl_xor(s, 16);
    s2 += __shfl_xor(s2, 16);
    if (half == 0) {
        partial[(size_t)blockIdx.x * COUT + o]          = s;
        partial[(size_t)(NBLK + blockIdx.x) * COUT + o] = s2;
    }
}

// GEMM1: A tile built on the fly = [points1 | IDW-interpolated points2].
// points1 columns go via async DMA (overlapped with the gather loop);
// points2 (16.8MB) is L2-resident; gathers are coalesced over channels.
__global__ void gemm1_fused_kernel(const float* __restrict__ points1,
                                   const float* __restrict__ points2,
                                   const int* __restrict__ knn_idx,
                                   const float* __restrict__ knn_w,
                                   const float* __restrict__ W,
                                   const float* __restrict__ bias,
                                   float* __restrict__ Hout,
                                   float* __restrict__ partial)
{
    constexpr int LDA = CIN1 + 4;                       // 388 floats: sA rows 16B-aligned
    __shared__ float sA[16 * LDA];
    __shared__ int   sIdx[16 * 3];
    __shared__ float sWt[16 * 3];

    const size_t mBase = (size_t)blockIdx.x * 16;       // 16 rows never straddle a batch
    const int b = (int)(mBase / N_);
    const int t = threadIdx.x;

    if (t < 48) {
        sIdx[t] = knn_idx[mBase * 3 + t];
        sWt[t]  = knn_w[mBase * 3 + t];
    }
    // points1 columns: 16 rows x 128 floats = 512 b128 chunks, one per thread
    {
        const int r  = t >> 5;                          // row 0..15
        const int ch = t & 31;                          // 16B chunk 0..31
        async_copy_b128(&sA[r * LDA + ch * 4],
                        points1 + (mBase + r) * C1_ + (size_t)ch * 4);
    }
    __syncthreads();                                     // sIdx/sWt visible
    const float* p2 = points2 + (size_t)b * S_ * C2_;    // interpolated columns
    for (int i = t; i < 16 * C2_; i += 512) {
        int r = i >> 8, c = i & 255;
        float v = sWt[r*3+0] * p2[(size_t)sIdx[r*3+0] * C2_ + c]
                + sWt[r*3+1] * p2[(size_t)sIdx[r*3+1] * C2_ + c]
                + sWt[r*3+2] * p2[(size_t)sIdx[r*3+2] * C2_ + c];
        sA[r * LDA + C1_ + c] = v;
    }
    wait_async0();                                       // drain this wave's DMA
    __syncthreads();                                     // whole tile visible
    gemm_core_and_epilogue<CIN1>(sA, W, bias, Hout, partial, mBase);
}

// GEMM2: A tile = BN(scale,shift)+ReLU applied to pre-BN h1 during staging
// (transform in flight -> must go through VGPRs, no async path here).
__global__ void gemm2_fused_kernel(const float* __restrict__ H1,
                                   const float* __restrict__ scale,
                                   const float* __restrict__ shift,
                                   const float* __restrict__ W,
                                   const float* __restrict__ bias,
                                   float* __restrict__ Hout,
                                   float* __restrict__ partial)
{
    constexpr int LDA = COUT + 4;
    __shared__ float sA[16 * LDA];
    const size_t mBase = (size_t)blockIdx.x * 16;
    const float* h1row = H1 + mBase * COUT;              // contiguous 16x256 tile
    for (int i = threadIdx.x; i < 16 * COUT; i += 512) {
        int r = i >> 8, c = i & 255;
        float v = h1row[i];
        sA[r * LDA + c] = fmaxf(v * scale[c] + shift[c], 0.0f);
    }
    __syncthreads();
    gemm_core_and_epilogue<COUT>(sA, W, bias, Hout, partial, mBase);
}

// ---------------------------------------------------------------------------
// Deterministic stats reduction: one block per channel; fixed strided
// accumulation + fixed LDS tree (no float atomics -> bitwise reproducible).
// scale = g*rsqrt(var+eps), shift = be - mean*scale  (biased var, ddof=0).
// ---------------------------------------------------------------------------
__global__ void reduce_stats_kernel(const float* __restrict__ partial,
                                    const float* __restrict__ gamma,
                                    const float* __restrict__ beta,
                                    float* __restrict__ scale,
                                    float* __restrict__ shift)
{
    __shared__ float red[256];
    const int c = blockIdx.x;
    const int t = threadIdx.x;
    float s = 0.f, s2 = 0.f;
    for (int i = t; i < NBLK; i += 256) {
        s  += partial[(size_t)i * COUT + c];
        s2 += partial[(size_t)(NBLK + i) * COUT + c];
    }
    red[t] = s;  __syncthreads();
    for (int step = 128; step > 0; step >>= 1) {
        if (t < step) red[t] += red[t + step];
        __syncthreads();
    }
    float sumAll = red[0]; __syncthreads();
    red[t] = s2; __syncthreads();
    for (int step = 128; step > 0; step >>= 1) {
        if (t < step) red[t] += red[t + step];
        __syncthreads();
    }
    if (t == 0) {
        float sqAll = red[0];
        float invM  = 1.0f / (float)M_;
        float mean  = sumAll * invM;
        float var   = sqAll * invM - mean * mean;
        float sc    = gamma[c] * rsqrtf(var + BN_EPS);
        scale[c] = sc;
        shift[c] = beta[c] - mean * sc;
    }
}

// Final layer-2 BN+ReLU in place on d_out; float4-vectorized (268MB pass).
// 256 threads handle 4 rows per block; thread t -> channels (t&63)*4 .. +3.
__global__ void bnrelu4_kernel(float4* __restrict__ H,
                               const float* __restrict__ scale,
                               const float* __restrict__ shift)
{
    const int t  = threadIdx.x;
    const int c4 = (t & 63) * 4;
    const size_t i = (size_t)blockIdx.x * 256 + t;       // float4 index, row-aligned
    float4 v = H[i];
    v.x = fmaxf(v.x * scale[c4 + 0] + shift[c4 + 0], 0.0f);
    v.y = fmaxf(v.y * scale[c4 + 1] + shift[c4 + 1], 0.0f);
    v.z = fmaxf(v.z * scale[c4 + 2] + shift[c4 + 2], 0.0f);
    v.w = fmaxf(v.w * scale[c4 + 3] + shift[c4 + 3], 0.0f);
    H[i] = v;
}

// ---------------------------------------------------------------------------
// d_ws layout (bytes, all 256-aligned):
//   [0)          knn_idx : M*3 ints    (1.50 MB)
//   [1572864)    knn_w   : M*3 floats  (1.50 MB)
//   [3145728)    partial : 2*NBLK*256 floats (16.78 MB)
//   [19922944)   scale|shift : 512 floats (reused by both layers; pipeline
//                is sequential on `stream`, so reuse is safe)
//   [19925248)   h1 (pre-BN) : M*256 floats (134.22 MB)
//   total ~154.2 MB.  d_out doubles as pre-BN h2, normalized in place.
// ---------------------------------------------------------------------------
extern "C" void kernel_launch(void* const* d_in, const int* in_sizes, int n_in,
                              void* d_out, int out_size, void* d_ws, size_t ws_size,
                              hipStream_t stream)
{
    const float* xyz1    = (const float*)d_in[0];
    const float* xyz2    = (const float*)d_in[1];
    const float* points1 = (const float*)d_in[2];
    const float* points2 = (const float*)d_in[3];
    const float* W1      = (const float*)d_in[4];
    const float* b1      = (const float*)d_in[5];
    const float* g1      = (const float*)d_in[6];
    const float* be1     = (const float*)d_in[7];
    const float* W2      = (const float*)d_in[8];
    const float* b2      = (const float*)d_in[9];
    const float* g2      = (const float*)d_in[10];
    const float* be2     = (const float*)d_in[11];

    char* ws = (char*)d_ws;
    int*   knn_idx = (int*)  (ws + 0);
    float* knn_w   = (float*)(ws + 1572864);
    float* partial = (float*)(ws + 3145728);
    float* scale   = (float*)(ws + 19922944);
    float* shift   = scale + 256;
    float* h1      = (float*)(ws + 19925248);
    float* h2      = (float*)d_out;

    // 1. 3-NN indices + weights
    knn_kernel<<<M_ / 256, 256, 0, stream>>>(xyz1, xyz2, knn_idx, knn_w);

    // 2. layer 1: fused interpolate+concat -> WMMA GEMM -> pre-BN h1 + partials
    gemm1_fused_kernel<<<NBLK, 512, 0, stream>>>(points1, points2, knn_idx,
                                                 knn_w, W1, b1, h1, partial);
    reduce_stats_kernel<<<COUT, 256, 0, stream>>>(partial, g1, be1, scale, shift);

    // 3. layer 2: BN+ReLU applied during A staging -> WMMA GEMM -> d_out + partials
    gemm2_fused_kernel<<<NBLK, 512, 0, stream>>>(h1, scale, shift,
                                                 W2, b2, h2, partial);
    reduce_stats_kernel<<<COUT, 256, 0, stream>>>(partial, g2, be2, scale, shift);

    // 4. final BN+ReLU in place on d_out (float4 path)
    bnrelu4_kernel<<<(M_ * COUT) / 1024, 256, 0, stream>>>((float4*)h2, scale, shift);
}